// focalTverskyLoss_55190329753766
// MI455X (gfx1250) — compile-verified
//
#include <hip/hip_runtime.h>

// Problem geometry (fixed by the reference).
#define HW      262144          // 512*512
#define NB      8
#define NC      10
#define NCLS    9
#define NPIX    (NB * HW)       // 2,097,152 pixels
#define CHUNKS  (NPIX / 4)      // 524,288 float4-chunks
#define BLOCKS  1024            // 8192 waves -> ~2x latency-hiding pool vs 512
#define THREADS 256             // 8 waves (wave32); 262144 threads -> exactly 2 chunks each
#define HW4     65536           // HW/4 : channel stride in float4 units

// -------- zero the 27 global counters (ws is poisoned each run) -----------
__global__ void ftl_init_kernel(unsigned* __restrict__ g) {
    if (threadIdx.x < 27) g[threadIdx.x] = 0u;
}

// -------- streaming argmax + packed per-lane histogram --------------------
// Per-lane counters: one u64 per histogram, 10 classes x 5-bit fields.
// Each lane sees exactly 8 pixels -> max field value 8 < 31, never overflows.
__global__ __launch_bounds__(THREADS) void ftl_hist_kernel(
        const float* __restrict__ logits,   // (8,10,512,512) f32
        const int*   __restrict__ targets,  // (8,512,512)    i32
        unsigned*    __restrict__ g)        // 27 counters: [t:9][p:9][tp:9]
{
    __shared__ unsigned long long lacc[THREADS * 3];
    __shared__ unsigned shfinal[27];
    if (threadIdx.x < 27) shfinal[threadIdx.x] = 0u;

    unsigned long long acc_t = 0ull, acc_p = 0ull, acc_tp = 0ull;

    const float4* __restrict__ in4 = (const float4*)logits;
    const int4*   __restrict__ tg4 = (const int4*)targets;

    const unsigned tid    = blockIdx.x * THREADS + threadIdx.x;
    const unsigned stride = BLOCKS * THREADS;   // 262144; CHUNKS/stride == 2 exactly

    for (unsigned i = tid; i < CHUNKS; i += stride) {
        const unsigned pix   = i << 2;
        const unsigned b     = pix >> 18;            // pix / HW
        const unsigned r     = pix & (HW - 1);       // pix % HW
        // float4 index of the channel-1 plane of this batch image
        const unsigned base4 = ((b * NC + 1) << 16) + (r >> 2);

        // Prefetch next iteration's first channel line (global_prefetch_b8).
        // Guard is wave-uniform: stride divides CHUNKS, lanes differ by <32.
        {
            const unsigned ni = i + stride;
            if (ni < CHUNKS) {
                const unsigned npix   = ni << 2;
                const unsigned nb     = npix >> 18;
                const unsigned nr     = npix & (HW - 1);
                const unsigned nbase4 = ((nb * NC + 1) << 16) + (nr >> 2);
                __builtin_prefetch((const void*)(in4 + nbase4), 0, 0);
            }
        }

        // ---- argmax over channels 1..9 (4 pixels per thread) ----
        // softmax is monotone -> argmax(logits) == argmax(softmax(logits))
        float4 v0 = in4[base4];
        float b0 = v0.x, b1 = v0.y, b2 = v0.z, b3 = v0.w;
        int   p0 = 1,    p1 = 1,    p2 = 1,    p3 = 1;
#pragma unroll
        for (int c = 1; c < NCLS; ++c) {
            const float4 v = in4[base4 + (unsigned)c * HW4];
            const int cls = c + 1;
            bool m;
            m = v.x > b0; b0 = m ? v.x : b0; p0 = m ? cls : p0;
            m = v.y > b1; b1 = m ? v.y : b1; p1 = m ? cls : p1;
            m = v.z > b2; b2 = m ? v.z : b2; p2 = m ? cls : p2;
            m = v.w > b3; b3 = m ? v.w : b3; p3 = m ? cls : p3;
        }

        const int4 t4 = tg4[i];
        const int tt[4] = { t4.x, t4.y, t4.z, t4.w };
        const int pp[4] = { p0, p1, p2, p3 };

        // ---- packed 5-bit-field counter updates (pure VALU, no ballots) ----
#pragma unroll
        for (int k = 0; k < 4; ++k) {
            const int t = tt[k];
            const int p = pp[k];
            const unsigned st = (unsigned)t * 5u;
            const unsigned sp = (unsigned)p * 5u;
            acc_t  += 1ull << st;                         // field 0 = ignored bg bin
            acc_p  += 1ull << ((t != 0) ? sp : 63u);      // bit-63 junk wraps, never read
            acc_tp += 1ull << ((t == p) ? st : 63u);      // t==p implies t>=1
        }
    }

    // ---- block reduction: spill packed counters, per-wave field sums ----
    const unsigned ti = threadIdx.x;
    lacc[ti * 3 + 0] = acc_t;
    lacc[ti * 3 + 1] = acc_p;
    lacc[ti * 3 + 2] = acc_tp;
    __syncthreads();

    const unsigned w = ti >> 5;        // wave id (0..7)
    const unsigned j = ti & 31u;       // lane id
    if (j < 27u) {
        const unsigned grp = j / 9u;           // 0=t, 1=p, 2=tp
        const unsigned cls = (j % 9u) + 1u;    // class 1..9
        const unsigned sh  = 5u * cls;
        unsigned s = 0u;
#pragma unroll
        for (int l = 0; l < 32; ++l) {
            // same-address reads within a lane-group broadcast for free
            s += (unsigned)((lacc[(w * 32u + (unsigned)l) * 3u + grp] >> sh) & 31ull);
        }
        atomicAdd(&shfinal[j], s);             // 8-way inter-wave merge, trivial
    }
    __syncthreads();
    if (ti < 27u) atomicAdd(&g[ti], shfinal[ti]);   // 27 global atomics per block
}

// -------- 9-class focal Tversky epilogue ----------------------------------
__global__ void ftl_final_kernel(const unsigned* __restrict__ g,
                                 float* __restrict__ out)
{
    if (threadIdx.x == 0 && blockIdx.x == 0) {
        float loss = 0.0f;
#pragma unroll
        for (int c = 0; c < NCLS; ++c) {
            const float T  = (float)g[c];
            const float P  = (float)g[9 + c];
            const float TP = (float)g[18 + c];
            const float FN = T - TP;
            const float FP = P - TP;
            const float tv = (TP + 1.0f) / (TP + 0.7f * FN + 0.3f * FP + 1.0f);
            loss += powf(1.0f - tv, 4.0f / 3.0f);
        }
        out[0] = loss;
    }
}

extern "C" void kernel_launch(void* const* d_in, const int* in_sizes, int n_in,
                              void* d_out, int out_size, void* d_ws, size_t ws_size,
                              hipStream_t stream) {
    const float* logits  = (const float*)d_in[0];   // (8,10,512,512) f32
    const int*   targets = (const int*)  d_in[1];   // (8,512,512)    i32
    unsigned*    g       = (unsigned*)d_ws;         // 27 u32 counters
    float*       out     = (float*)d_out;           // scalar loss

    ftl_init_kernel <<<1, 32, 0, stream>>>(g);
    ftl_hist_kernel <<<BLOCKS, THREADS, 0, stream>>>(logits, targets, g);
    ftl_final_kernel<<<1, 32, 0, stream>>>(g, out);
}